// MambaBlock_42322607735163
// MI455X (gfx1250) — compile-verified
//
#include <hip/hip_runtime.h>
#include <math.h>

// ---------------------------------------------------------------------------
// Mamba block forward for MI455X (gfx1250), wave32.
// GEMMs: V_WMMA_F32_16X16X4_F32 (full fp32, matrix pipe), 32x32 wave tiles,
// double-buffered LDS fed by GLOBAL_LOAD_ASYNC_TO_LDS_B128 (ASYNCcnt path)
// via inline asm (portable across toolchains per CDNA5_HIP.md), else
// synchronous staging on non-gfx1250 compiles.
// ---------------------------------------------------------------------------

typedef __attribute__((ext_vector_type(2))) float v2f;
typedef __attribute__((ext_vector_type(8))) float v8f;

#define D_MODEL 1024
#define D_STATE 16
#define D_CONV  4
#define D_INNER 2048
#define BATCH   2
#define SEQ     1024
#define NTOK    (BATCH * SEQ)   // 2048 tokens

#if defined(__gfx1250__)
#define HAVE_ASYNC 1
#else
#define HAVE_ASYNC 0
#endif

__device__ __forceinline__ float sigmoidf_(float x) { return 1.0f / (1.0f + __expf(-x)); }

#if HAVE_ASYNC
// 16B global -> LDS direct copy, no VGPR data staging, tracked by ASYNCcnt.
// VDST = LDS byte address (flat LDS pointer low 32 bits == LDS offset,
// ISA 10.2 aperture mapping); VADDR = 64-bit global address (VGPR pair).
__device__ __forceinline__ void async_copy16(const float* g, float* l) {
  asm volatile("global_load_async_to_lds_b128 %0, %1, off"
               :: "v"((unsigned int)(uintptr_t)l), "v"(g)
               : "memory");
}
#define GEMM_WAIT_ASYNC(n) asm volatile("s_wait_asynccnt %0" ::"i"(n) : "memory")
#endif

// ---------------------------------------------------------------------------
// LayerNorm over D_MODEL=1024 per token. One 256-thread block per token.
// ---------------------------------------------------------------------------
__launch_bounds__(256)
__global__ void ln_kernel(const float* __restrict__ x, const float* __restrict__ g,
                          const float* __restrict__ b, float* __restrict__ xn) {
  const int tok = blockIdx.x;
  const int tid = threadIdx.x;
  const float* xr = x + (size_t)tok * D_MODEL;
  float v[4];
  float s = 0.f, s2 = 0.f;
#pragma unroll
  for (int i = 0; i < 4; ++i) {
    v[i] = xr[tid + 256 * i];
    s += v[i]; s2 += v[i] * v[i];
  }
#pragma unroll
  for (int off = 16; off > 0; off >>= 1) {
    s  += __shfl_xor(s,  off, 32);
    s2 += __shfl_xor(s2, off, 32);
  }
  __shared__ float rs[8], rq[8];
  if ((tid & 31) == 0) { rs[tid >> 5] = s; rq[tid >> 5] = s2; }
  __syncthreads();
  float ts = 0.f, tq = 0.f;
#pragma unroll
  for (int i = 0; i < 8; ++i) { ts += rs[i]; tq += rq[i]; }
  const float mu  = ts * (1.0f / D_MODEL);
  const float var = tq * (1.0f / D_MODEL) - mu * mu;
  const float inv = rsqrtf(var + 1e-5f);
  float* yr = xn + (size_t)tok * D_MODEL;
#pragma unroll
  for (int i = 0; i < 4; ++i) {
    const int c = tid + 256 * i;
    yr[c] = (v[i] - mu) * inv * g[c] + b[c];
  }
}

// ---------------------------------------------------------------------------
// Staging of one K-chunk (32) into LDS buffers (A: 128x32, B: 32x64).
// LDS row strides 36 / 72 floats: 16B-aligned rows (b128-legal) and
// conflict-free fragment reads (gcd analysis vs 64 banks).
// ---------------------------------------------------------------------------
__device__ __forceinline__ void stage_chunk(const float* __restrict__ A,
                                            const float* __restrict__ Bw,
                                            int M, int N, int K, int m0, int n0,
                                            int kc, int tid,
                                            float (*__restrict__ Asb)[36],
                                            float (*__restrict__ Bsb)[72]) {
  // ---- A: 128x32 floats = 1024 B128 slots, 4 per thread ----
#pragma unroll
  for (int it = 0; it < 4; ++it) {
    const int slot = tid + 256 * it;
    const int row = slot >> 3, col = (slot & 7) << 2;
    const float* gp = A + (size_t)(m0 + row) * K + kc + col;
#if HAVE_ASYNC
    if (m0 + row < M) {
      async_copy16(gp, &Asb[row][col]);
    } else {
      Asb[row][col + 0] = 0.f; Asb[row][col + 1] = 0.f;
      Asb[row][col + 2] = 0.f; Asb[row][col + 3] = 0.f;
    }
#else
    float4 v = make_float4(0.f, 0.f, 0.f, 0.f);
    if (m0 + row < M) v = *(const float4*)gp;
    Asb[row][col + 0] = v.x; Asb[row][col + 1] = v.y;
    Asb[row][col + 2] = v.z; Asb[row][col + 3] = v.w;
#endif
  }
  // ---- B: 32x64 floats = 512 B128 slots, 2 per thread ----
#pragma unroll
  for (int it = 0; it < 2; ++it) {
    const int slot = tid + 256 * it;
    const int row = slot >> 4, col = (slot & 15) << 2;
    const float* gp = Bw + (size_t)(kc + row) * N + n0 + col;
#if HAVE_ASYNC
    if (n0 + col + 3 < N) {
      async_copy16(gp, &Bsb[row][col]);
    } else {
      for (int j = 0; j < 4; ++j)
        Bsb[row][col + j] = (n0 + col + j < N) ? gp[j] : 0.f;
    }
#else
    if (n0 + col + 3 < N) {
      float4 v = *(const float4*)gp;
      Bsb[row][col + 0] = v.x; Bsb[row][col + 1] = v.y;
      Bsb[row][col + 2] = v.z; Bsb[row][col + 3] = v.w;
    } else {
      for (int j = 0; j < 4; ++j)
        Bsb[row][col + j] = (n0 + col + j < N) ? gp[j] : 0.f;
    }
#endif
  }
}

// ---------------------------------------------------------------------------
// GEMM: C[M,N] = epilogue(A[M,K] @ Bw[K,N]).
//   act==1 : softplus after optional bias (delta path)
//   add    : optional residual added at store (output projection)
// 256 threads = 8 waves (4x2); block tile 128x64; wave tile 32x32
// (4 WMMA accumulators). K chunked by 32, double-buffered LDS.
// f32 16x16x4 operand layouts (ISA 7.12.2):
//   A 16x4 : lanes 0-15 hold M=lane, VGPR{0,1}=K{0,1}; lanes 16-31: K{2,3}
//   B 4x16 : VGPR i, lanes 0-15 -> K=i, lanes 16-31 -> K=2+i; col = lane&15
//   C/D    : VGPR i, lanes 0-15 -> M=i, lanes 16-31 -> M=8+i; col = lane&15
// ---------------------------------------------------------------------------
__launch_bounds__(256)
__global__ void gemm_wmma_kernel(const float* __restrict__ A, const float* __restrict__ Bw,
                                 const float* __restrict__ bias, const float* __restrict__ add,
                                 float* __restrict__ C, int M, int N, int K, int act) {
  __shared__ float As[2][128][36];   // 2 x 18 KB
  __shared__ float Bs[2][32][72];    // 2 x  9 KB

  const int tid  = threadIdx.x;
  const int m0   = blockIdx.y * 128;
  const int n0   = blockIdx.x * 64;
  const int wid  = tid >> 5;
  const int lane = tid & 31;
  const int wrow = (wid & 3) * 32;   // wave's 32-row stripe in block tile
  const int wcol = (wid >> 2) * 32;  // wave's 32-col stripe
  const int lh   = lane >> 4;        // lane half
  const int l15  = lane & 15;

  v8f acc[2][2];
#pragma unroll
  for (int rt = 0; rt < 2; ++rt)
#pragma unroll
    for (int ct = 0; ct < 2; ++ct) acc[rt][ct] = (v8f){};

  const int nch = K >> 5;
  // prologue: chunk 0 into buffer 0
  stage_chunk(A, Bw, M, N, K, m0, n0, 0, tid, As[0], Bs[0]);

  for (int ic = 0; ic < nch; ++ic) {
    const int cur = ic & 1;
    const bool more = (ic + 1) < nch;
    if (more)  // overlap: issue next chunk into other buffer
      stage_chunk(A, Bw, M, N, K, m0, n0, (ic + 1) << 5, tid, As[cur ^ 1], Bs[cur ^ 1]);
#if HAVE_ASYNC
    // in-order ASYNC completion: <=6 outstanding means current chunk landed
    if (more) GEMM_WAIT_ASYNC(6); else GEMM_WAIT_ASYNC(0);
#endif
    __syncthreads();

#pragma unroll
    for (int k = 0; k < 32; k += 4) {
      v2f a0, a1;
      a0.x = As[cur][wrow + l15][k + 2 * lh + 0];
      a0.y = As[cur][wrow + l15][k + 2 * lh + 1];
      a1.x = As[cur][wrow + 16 + l15][k + 2 * lh + 0];
      a1.y = As[cur][wrow + 16 + l15][k + 2 * lh + 1];
      v2f b0, b1;
      b0.x = Bs[cur][k + 2 * lh + 0][wcol + l15];
      b0.y = Bs[cur][k + 2 * lh + 1][wcol + l15];
      b1.x = Bs[cur][k + 2 * lh + 0][wcol + 16 + l15];
      b1.y = Bs[cur][k + 2 * lh + 1][wcol + 16 + l15];
      acc[0][0] = __builtin_amdgcn_wmma_f32_16x16x4_f32(false, a0, false, b0,
                                                        (short)0, acc[0][0], false, false);
      acc[0][1] = __builtin_amdgcn_wmma_f32_16x16x4_f32(false, a0, false, b1,
                                                        (short)0, acc[0][1], false, false);
      acc[1][0] = __builtin_amdgcn_wmma_f32_16x16x4_f32(false, a1, false, b0,
                                                        (short)0, acc[1][0], false, false);
      acc[1][1] = __builtin_amdgcn_wmma_f32_16x16x4_f32(false, a1, false, b1,
                                                        (short)0, acc[1][1], false, false);
    }
    __syncthreads();   // all waves done reading buffer `cur`
  }

  // ---- epilogue + store (C/D layout: VGPR i -> M = i + 8*laneHalf) ----
#pragma unroll
  for (int rt = 0; rt < 2; ++rt) {
    const int gm0 = m0 + wrow + 16 * rt + 8 * lh;
#pragma unroll
    for (int ct = 0; ct < 2; ++ct) {
      const int gn = n0 + wcol + 16 * ct + l15;
      if (gn >= N) continue;
#pragma unroll
      for (int i = 0; i < 8; ++i) {
        const int gm = gm0 + i;
        if (gm >= M) continue;
        float v = acc[rt][ct][i];
        if (bias) v += bias[gn];
        if (act == 1) v = (v > 20.f) ? v : log1pf(__expf(v));  // softplus
        if (add) v += add[(size_t)gm * N + gn];
        C[(size_t)gm * N + gn] = v;
      }
    }
  }
}

// ---------------------------------------------------------------------------
// Causal depthwise conv1d (K=4) + bias + SiLU. Input is first half of xz.
// ---------------------------------------------------------------------------
__launch_bounds__(256)
__global__ void conv_silu_kernel(const float* __restrict__ xz, const float* __restrict__ w,
                                 const float* __restrict__ cb, float* __restrict__ xb) {
  const int idx = blockIdx.x * blockDim.x + threadIdx.x;
  if (idx >= NTOK * D_INNER) return;
  const int d   = idx & (D_INNER - 1);
  const int tok = idx >> 11;              // D_INNER = 2048
  const int l   = tok & (SEQ - 1);
  const int bb  = tok >> 10;              // SEQ = 1024
  float acc = cb[d];
#pragma unroll
  for (int j = 0; j < D_CONV; ++j) {
    const int ll = l - (D_CONV - 1) + j;
    if (ll >= 0)
      acc += w[d * D_CONV + j] * xz[(size_t)(bb * SEQ + ll) * (2 * D_INNER) + d];
  }
  xb[(size_t)tok * D_INNER + d] = acc * sigmoidf_(acc);
}

// ---------------------------------------------------------------------------
// Selective scan. 2 chains per wave32: lane = state s (0..15) + 16*chainHalf.
// h[s] = exp(delta*A[d,s])*h[s] + delta*B[t,s]*xb ; y = sum_s h[s]*C[t,s]
// 16-lane shfl_xor butterfly for the state reduction.
// ---------------------------------------------------------------------------
__launch_bounds__(256)
__global__ void scan_kernel(const float* __restrict__ delta, const float* __restrict__ xb,
                            const float* __restrict__ Bm, const float* __restrict__ Cm,
                            const float* __restrict__ logA, float* __restrict__ y) {
  const int lane  = threadIdx.x & 31;
  const int gw    = (blockIdx.x * blockDim.x + threadIdx.x) >> 5;
  const int chain = gw * 2 + (lane >> 4);      // chains = BATCH*D_INNER = 4096 exactly
  const int d  = chain & (D_INNER - 1);
  const int bb = chain >> 11;
  const int s  = lane & 15;
  const float An = -__expf(logA[d * D_STATE + s]);
  const size_t base = (size_t)bb * SEQ;
  const float* dl = delta + base * D_INNER + d;
  const float* xl = xb    + base * D_INNER + d;
  const float* Bl = Bm + base * D_STATE + s;
  const float* Cl = Cm + base * D_STATE + s;
  float* yl = y + base * D_INNER + d;
  float h = 0.f;
  for (int t = 0; t < SEQ; ++t) {
    const float dv = dl[(size_t)t * D_INNER];
    const float xv = xl[(size_t)t * D_INNER];
    const float a  = __expf(dv * An);
    h = a * h + dv * Bl[t * D_STATE] * xv;
    float p = h * Cl[t * D_STATE];
    p += __shfl_xor(p, 1, 32);
    p += __shfl_xor(p, 2, 32);
    p += __shfl_xor(p, 4, 32);
    p += __shfl_xor(p, 8, 32);
    if (s == 0) yl[(size_t)t * D_INNER] = p;
  }
}

// ---------------------------------------------------------------------------
// Gate: y = (y_scan + xb * Dp) * silu(z);  z is the second half of xz.
// ---------------------------------------------------------------------------
__launch_bounds__(256)
__global__ void gate_kernel(const float* __restrict__ xz, const float* __restrict__ xb,
                            const float* __restrict__ Dp, float* __restrict__ y) {
  const int idx = blockIdx.x * blockDim.x + threadIdx.x;
  if (idx >= NTOK * D_INNER) return;
  const int d   = idx & (D_INNER - 1);
  const int tok = idx >> 11;
  const float z  = xz[(size_t)tok * (2 * D_INNER) + D_INNER + d];
  const float yv = y[idx] + xb[idx] * Dp[d];
  y[idx] = yv * (z * sigmoidf_(z));
}

// ---------------------------------------------------------------------------
// Host launch
// ---------------------------------------------------------------------------
extern "C" void kernel_launch(void* const* d_in, const int* in_sizes, int n_in,
                              void* d_out, int out_size, void* d_ws, size_t ws_size,
                              hipStream_t stream) {
  const float* x      = (const float*)d_in[0];
  const float* ln_g   = (const float*)d_in[1];
  const float* ln_b   = (const float*)d_in[2];
  const float* W_in   = (const float*)d_in[3];
  const float* conv_w = (const float*)d_in[4];
  const float* conv_b = (const float*)d_in[5];
  const float* W_b    = (const float*)d_in[6];
  const float* W_c    = (const float*)d_in[7];
  const float* W_d    = (const float*)d_in[8];
  const float* b_d    = (const float*)d_in[9];
  const float* logA   = (const float*)d_in[10];
  const float* Dp     = (const float*)d_in[11];
  const float* W_out  = (const float*)d_in[12];
  float* out = (float*)d_out;

  // workspace layout (~88.3 MB of fp32)
  float* ws  = (float*)d_ws;
  float* xn  = ws;                               // NTOK * D_MODEL
  float* xz  = xn  + (size_t)NTOK * D_MODEL;     // NTOK * 2*D_INNER
  float* xb  = xz  + (size_t)NTOK * 2 * D_INNER; // NTOK * D_INNER
  float* dl  = xb  + (size_t)NTOK * D_INNER;     // NTOK * D_INNER
  float* Bmv = dl  + (size_t)NTOK * D_INNER;     // NTOK * D_STATE
  float* Cmv = Bmv + (size_t)NTOK * D_STATE;     // NTOK * D_STATE
  float* yv  = Cmv + (size_t)NTOK * D_STATE;     // NTOK * D_INNER (gated in-place)

  const dim3 blk(256);
  const int n_el = NTOK * D_INNER;

  // 1. LayerNorm
  ln_kernel<<<NTOK, blk, 0, stream>>>(x, ln_g, ln_b, xn);
  // 2. xz = xn @ W_in  (2048 x 1024 x 4096)
  gemm_wmma_kernel<<<dim3((2 * D_INNER) / 64, NTOK / 128), blk, 0, stream>>>(
      xn, W_in, nullptr, nullptr, xz, NTOK, 2 * D_INNER, D_MODEL, 0);
  // 3. causal depthwise conv + SiLU -> xb
  conv_silu_kernel<<<(n_el + 255) / 256, blk, 0, stream>>>(xz, conv_w, conv_b, xb);
  // 4. delta = softplus(xb @ W_delta + b_delta)
  gemm_wmma_kernel<<<dim3(D_INNER / 64, NTOK / 128), blk, 0, stream>>>(
      xb, W_d, b_d, nullptr, dl, NTOK, D_INNER, D_INNER, 1);
  // 5. Bm = xb @ W_b ; Cm = xb @ W_c  (N = 16)
  gemm_wmma_kernel<<<dim3(1, NTOK / 128), blk, 0, stream>>>(
      xb, W_b, nullptr, nullptr, Bmv, NTOK, D_STATE, D_INNER, 0);
  gemm_wmma_kernel<<<dim3(1, NTOK / 128), blk, 0, stream>>>(
      xb, W_c, nullptr, nullptr, Cmv, NTOK, D_STATE, D_INNER, 0);
  // 6. selective scan -> yv
  scan_kernel<<<256, blk, 0, stream>>>(dl, xb, Bmv, Cmv, logA, yv);
  // 7. gate: yv = (yv + xb*Dp) * silu(z)
  gate_kernel<<<(n_el + 255) / 256, blk, 0, stream>>>(xz, xb, Dp, yv);
  // 8. out = yv @ W_out + residual(x)
  gemm_wmma_kernel<<<dim3(D_MODEL / 64, NTOK / 128), blk, 0, stream>>>(
      yv, W_out, nullptr, x, out, NTOK, D_MODEL, D_INNER, 0);
}